// onlyCrf_9182640079408
// MI455X (gfx1250) — compile-verified
//
#include <hip/hip_runtime.h>

typedef float v2f __attribute__((ext_vector_type(2)));
typedef float v8f __attribute__((ext_vector_type(8)));

constexpr int Bn = 64, Sn = 512, Hn = 768, Tn = 16;
constexpr int KSTEPS = Hn / 4;   // 192 WMMA k-steps of 4

// ---------------------------------------------------------------------------
// Kernel 1: fused embedding gather + (32768x768)x(768x16) GEMM via
// V_WMMA_F32_16X16X4_F32. One wave computes one 16-token x 16-tag tile.
// W is staged into LDS in the exact WMMA B-matrix lane layout.
// ---------------------------------------------------------------------------
__global__ __launch_bounds__(256) void embed_gemm_wmma(
    const int*   __restrict__ src,       // [B,S]
    const float* __restrict__ emb,       // [V,H]
    const float* __restrict__ W,         // [H,T]
    const float* __restrict__ bias,      // [T]
    float*       __restrict__ emissions) // [S,B,T]
{
    __shared__ float Wlds[KSTEPS * 32 * 2];   // 48 KB

    const int tid  = threadIdx.x;
    const int lane = tid & 31;
    const int half = lane >> 4;   // 0: K+{0,1}, 1: K+{2,3}
    const int n    = lane & 15;   // N (tag) column / A row

    // Stage W into LDS in B-matrix layout: VGPR0 holds K=ks*4+2*half+0,
    // VGPR1 holds K=ks*4+2*half+1, column = lane&15.
    for (int idx = tid; idx < KSTEPS * 32; idx += 256) {
        int ks = idx >> 5;
        int l  = idx & 31;
        int k  = ks * 4 + (l >> 4) * 2;
        int nn = l & 15;
        Wlds[idx * 2 + 0] = W[(k + 0) * Tn + nn];
        Wlds[idx * 2 + 1] = W[(k + 1) * Tn + nn];
    }
    __syncthreads();

    const int tile = blockIdx.x * 8 + (tid >> 5);   // 2048 tiles of 16 tokens
    const int tok  = tile * 16 + n;                  // A-matrix row = lane&15
    const int vidx = src[tok];
    const float* rowptr = emb + (size_t)vidx * Hn + half * 2;

    v8f acc = {0.f, 0.f, 0.f, 0.f, 0.f, 0.f, 0.f, 0.f};
    const v2f* wl = (const v2f*)Wlds;

#pragma unroll 4
    for (int ks = 0; ks < KSTEPS; ++ks) {
        v2f a    = *(const v2f*)(rowptr + ks * 4);   // emb row, K pair (8B aligned)
        v2f bmat = wl[ks * 32 + lane];               // W tile, B layout
        acc = __builtin_amdgcn_wmma_f32_16x16x4_f32(false, a, false, bmat,
                                                    (short)0, acc, false, false);
    }

    const float bv = bias[n];
#pragma unroll
    for (int r = 0; r < 8; ++r) {
        int mrow = r + half * 8;           // C/D layout: VGPR r -> rows r, r+8
        int t2   = tile * 16 + mrow;
        int bb   = t2 >> 9;                // token -> (b, s), tokens ordered b*S+s
        int ss   = t2 & 511;
        emissions[(ss * Bn + bb) * Tn + n] = acc[r] + bv;
    }
}

// ---------------------------------------------------------------------------
// Kernel 2: CRF numerator + forward (logsumexp) + Viterbi, one wave per batch
// row. Lanes j (mirrored in upper half-wave) hold alpha_j / score_j; the
// transition column trans[:,j] lives in 16 registers per lane. History in LDS.
// ---------------------------------------------------------------------------
__global__ __launch_bounds__(32) void crf_scan(
    const float* __restrict__ emissions, // [S,B,T]
    const int*   __restrict__ label,     // [B,S]
    const int*   __restrict__ pmask,     // [B,S]
    const float* __restrict__ start_t,   // [T]
    const float* __restrict__ end_t,     // [T]
    const float* __restrict__ trans,     // [T,T]
    int*         __restrict__ pred,      // [B,S]
    float*       __restrict__ ll)        // [B]
{
    __shared__ unsigned char hist[Sn * Tn];   // 8 KB backpointers

    const int b    = blockIdx.x;
    const int lane = threadIdx.x;
    const int j    = lane & 15;

    float tcol[Tn];
#pragma unroll
    for (int i = 0; i < Tn; ++i) tcol[i] = trans[i * Tn + j];
    const float endj = end_t[j];

    float em     = emissions[(0 * Bn + b) * Tn + j];
    float alpha  = start_t[j] + em;
    float vscore = alpha;

    int   tagp = label[b * Sn + 0];
    float num  = 0.f;
    int   slen = 0;
    if (lane == 0) {
        num  = start_t[tagp] + emissions[(0 * Bn + b) * Tn + tagp];
        slen = pmask[b * Sn + 0];
    }

    for (int s = 1; s < Sn; ++s) {
        em = emissions[(s * Bn + b) * Tn + j];
        int   mi = pmask[b * Sn + s];

        // forward algorithm step: nxt_j = LSE_i(alpha_i + trans[i][j]) + em_j
        float t[Tn];
        float mmax = -3.402823466e38f;
#pragma unroll
        for (int i = 0; i < Tn; ++i) {
            float ai = __shfl(alpha, i, 32);
            t[i] = ai + tcol[i];
            mmax = fmaxf(mmax, t[i]);
        }
        float ssum = 0.f;
#pragma unroll
        for (int i = 0; i < Tn; ++i) ssum += __expf(t[i] - mmax);
        float nxt = mmax + __logf(ssum) + em;
        alpha = (mi > 0) ? nxt : alpha;

        // Viterbi step (full-length, no mask, per reference)
        float best = -3.402823466e38f;
        int   bi   = 0;
#pragma unroll
        for (int i = 0; i < Tn; ++i) {
            float v = __shfl(vscore, i, 32) + tcol[i];
            if (v > best) { best = v; bi = i; }   // first-max tie rule == argmax
        }
        vscore = best + em;
        if (lane < 16) hist[s * Tn + j] = (unsigned char)bi;

        // gold-path numerator (lane 0)
        if (lane == 0) {
            int tc = label[b * Sn + s];
            num += (trans[tagp * Tn + tc] + emissions[(s * Bn + b) * Tn + tc]) * (float)mi;
            tagp = tc;
            slen += mi;
        }
    }

    // denominator: logsumexp over lanes of alpha + end
    float x  = alpha + endj;
    float mx = x;
#pragma unroll
    for (int off = 8; off >= 1; off >>= 1) mx = fmaxf(mx, __shfl_xor(mx, off, 32));
    float se = __expf(x - mx);
#pragma unroll
    for (int off = 8; off >= 1; off >>= 1) se += __shfl_xor(se, off, 32);
    float den = mx + __logf(se);

    // Viterbi terminal argmax (lowest index on ties)
    float y  = vscore + endj;
    int   bj = j;
#pragma unroll
    for (int off = 8; off >= 1; off >>= 1) {
        float oy = __shfl_xor(y, off, 32);
        int   oj = __shfl_xor(bj, off, 32);
        if (oy > y || (oy == y && oj < bj)) { y = oy; bj = oj; }
    }

    __syncthreads();   // hist (LDS) visible for backtrack

    if (lane == 0) {
        int lt = label[b * Sn + (slen - 1)];
        num += end_t[lt];
        ll[b] = num - den;

        int tag = bj;
        pred[b * Sn + (Sn - 1)] = tag;
        for (int s = Sn - 1; s >= 1; --s) {
            tag = (int)hist[s * Tn + tag];
            pred[b * Sn + (s - 1)] = tag;
        }
    }
}

// ---------------------------------------------------------------------------
// Kernel 3: outputs. out = [loss, correct, pred_flat(32768), label_flat(32768)]
// ---------------------------------------------------------------------------
__global__ __launch_bounds__(1024) void finalize(
    const int*   __restrict__ label,
    const int*   __restrict__ pred,
    const float* __restrict__ ll,
    float*       __restrict__ out)
{
    __shared__ int scnt;
    if (threadIdx.x == 0) scnt = 0;
    __syncthreads();

    const int N = Bn * Sn;
    int local = 0;
    for (int i = threadIdx.x; i < N; i += 1024) {
        int p   = pred[i];
        int lab = label[i];
        int pf  = (lab > 0) ? p : 0;
        out[2 + i]     = (float)pf;
        out[2 + N + i] = (float)lab;
        local += (pf == lab) ? 1 : 0;
    }
    atomicAdd(&scnt, local);
    __syncthreads();

    if (threadIdx.x == 0) {
        float s = 0.f;
        for (int bb = 0; bb < Bn; ++bb) s += ll[bb];
        out[0] = -s / (float)Bn;
        out[1] = (float)scnt;
    }
}

// ---------------------------------------------------------------------------
extern "C" void kernel_launch(void* const* d_in, const int* in_sizes, int n_in,
                              void* d_out, int out_size, void* d_ws, size_t ws_size,
                              hipStream_t stream) {
    const int*   src   = (const int*)d_in[0];
    const int*   label = (const int*)d_in[1];
    const int*   pmask = (const int*)d_in[2];
    const float* emb   = (const float*)d_in[3];
    const float* W     = (const float*)d_in[4];
    const float* bias  = (const float*)d_in[5];
    const float* st    = (const float*)d_in[6];
    const float* en    = (const float*)d_in[7];
    const float* tr    = (const float*)d_in[8];
    float* out = (float*)d_out;

    char*  ws        = (char*)d_ws;
    float* emissions = (float*)ws;                               // S*B*T f32 = 2 MB
    int*   pred      = (int*)(ws + (size_t)Sn * Bn * Tn * 4);    // B*S i32
    float* ll        = (float*)(ws + (size_t)Sn * Bn * Tn * 4
                                   + (size_t)Bn * Sn * 4);       // B f32

    embed_gemm_wmma<<<256, 256, 0, stream>>>(src, emb, W, bias, emissions);
    crf_scan<<<Bn, 32, 0, stream>>>(emissions, label, pmask, st, en, tr, pred, ll);
    finalize<<<1, 1024, 0, stream>>>(label, pred, ll, out);
}